// TransformerBlockQuantum_65481071406373
// MI455X (gfx1250) — compile-verified
//
#include <hip/hip_runtime.h>

typedef __attribute__((ext_vector_type(2))) float v2f;
typedef __attribute__((ext_vector_type(8))) float v8f;

#define EMBED 1024
#define FFN   4096
#define NTOK  (8 * 2048)

// ---------------------------------------------------------------------------
// Kernel 0: compute circuit expvals -> head vector (1024) and relu(W2 cvec_f)
// (4096). Each thread recomputes the 8 cosines (trivial) to avoid a launch.
// ---------------------------------------------------------------------------
__global__ __launch_bounds__(256) void k_const_small(
    const float* __restrict__ attn_params, const float* __restrict__ ffn_params,
    const float* __restrict__ w2, float* __restrict__ hvec,
    float* __restrict__ relu_vec)
{
    const int f = blockIdx.x * 256 + threadIdx.x;   // 0..4095

    float a0 = cosf(attn_params[0]), a1 = cosf(attn_params[1]);
    float a2 = cosf(attn_params[2]), a3 = cosf(attn_params[3]);
    float cv0 = a1 * a2 * a3;
    float cv1 = a0 * a1;
    float cv2 = a0 * a1 * a2;
    float cv3 = a0 * a1 * a2 * a3;

    float f0 = cosf(ffn_params[0]), f1 = cosf(ffn_params[1]);
    float f2 = cosf(ffn_params[2]), f3 = cosf(ffn_params[3]);
    float cf0 = f1 * f2 * f3;
    float cf1 = f0 * f1;
    float cf2 = f0 * f1 * f2;
    float cf3 = f0 * f1 * f2 * f3;

    float t = w2[f * 4 + 0] * cf0 + w2[f * 4 + 1] * cf1 +
              w2[f * 4 + 2] * cf2 + w2[f * 4 + 3] * cf3;
    relu_vec[f] = t > 0.f ? t : 0.f;

    if (f < EMBED) {
        int r = f & 3;
        hvec[f] = (r == 0) ? cv0 : (r == 1) ? cv1 : (r == 2) ? cv2 : cv3;
    }
}

// ---------------------------------------------------------------------------
// Kernel 1: constant mat-vec via V_WMMA_F32_16X16X4_F32 (full fp32).
// Blocks 0..63   : attn_out[16m .. 16m+15] = W_mix(1024x1024) . hvec
// Blocks 64..127 : ffn_out [16m .. 16m+15] = W_out(1024x4096) . relu_vec
// B matrix = h-chunk broadcast to all 16 columns -> every D column equals the
// mat-vec partial; k-slot ordering cancels between A and B fills.
// One wave per block; EXEC all ones as WMMA requires.
// ---------------------------------------------------------------------------
__global__ __launch_bounds__(32) void k_matvec_wmma(
    const float* __restrict__ w_mix, const float* __restrict__ w_out,
    const float* __restrict__ hvec, const float* __restrict__ relu_vec,
    float* __restrict__ attn_out, float* __restrict__ ffn_out)
{
    const int lane = threadIdx.x;
    const int half = lane >> 4;     // 0: k slots {0,1}, 1: k slots {2,3}
    const int lm   = lane & 15;     // row within 16-row block

    const bool is_ffn = blockIdx.x >= 64;
    const int  m = is_ffn ? (int)blockIdx.x - 64 : (int)blockIdx.x;
    const int  K = is_ffn ? FFN : EMBED;

    const float* W = (is_ffn ? w_out : w_mix) + (size_t)(m * 16 + lm) * K + half * 2;
    const float* h = (is_ffn ? relu_vec : hvec) + half * 2;

    v8f acc = {0.f, 0.f, 0.f, 0.f, 0.f, 0.f, 0.f, 0.f};

    const int nkb = K >> 2;
    for (int kb = 0; kb < nkb; ++kb) {
        v2f a, b;
        a.x = W[kb * 4 + 0];
        a.y = W[kb * 4 + 1];
        b.x = h[kb * 4 + 0];
        b.y = h[kb * 4 + 1];
        acc = __builtin_amdgcn_wmma_f32_16x16x4_f32(
            /*neg_a=*/false, a, /*neg_b=*/false, b,
            /*c_mod=*/(short)0, acc, /*reuse_a=*/false, /*reuse_b=*/false);
    }

    // D layout: VGPR r holds M=r (lanes 0-15) / M=8+r (lanes 16-31), N=lane%16.
    // Column 0 (lanes 0 and 16) carries the mat-vec result.
    if (lm == 0) {
        float* o = (is_ffn ? ffn_out : attn_out) + m * 16 + half * 8;
#pragma unroll
        for (int r = 0; r < 8; ++r) o[r] = acc[r];
    }
}

// ---------------------------------------------------------------------------
// Kernel 2: fused  out = LN2( LN1(x + attn_out) + ffn_out ).
// One wave32 per token: 32 lanes x 8 float4 = 1024 floats held in registers.
// Reductions via 5 shfl_xor steps; no LDS, no barriers. Memory-optimal:
// read x once, write out once (128 MiB total -> ~5.5us at 23.3 TB/s).
// ---------------------------------------------------------------------------
__global__ __launch_bounds__(256) void k_fused_ln(
    const float* __restrict__ x, const float* __restrict__ attn_out,
    const float* __restrict__ ffn_out,
    const float* __restrict__ g1, const float* __restrict__ b1,
    const float* __restrict__ g2, const float* __restrict__ b2,
    float* __restrict__ out)
{
    const int lane = threadIdx.x & 31;
    const int wave = threadIdx.x >> 5;
    const size_t token = (size_t)blockIdx.x * 8 + wave;

    const float4* xr   = (const float4*)x + token * 256;
    float4*       orow = (float4*)out + token * 256;
    const float4* at4  = (const float4*)attn_out;
    const float4* ff4  = (const float4*)ffn_out;
    const float4* g14  = (const float4*)g1;
    const float4* b14  = (const float4*)b1;
    const float4* g24  = (const float4*)g2;
    const float4* b24  = (const float4*)b2;

    float4 v[8];
    float s = 0.f, ss = 0.f;
#pragma unroll
    for (int i = 0; i < 8; ++i) {
        const int idx = i * 32 + lane;
        float4 xv = xr[idx];
        float4 av = at4[idx];
        xv.x += av.x; xv.y += av.y; xv.z += av.z; xv.w += av.w;
        v[i] = xv;
        s  += xv.x + xv.y + xv.z + xv.w;
        ss += xv.x * xv.x + xv.y * xv.y + xv.z * xv.z + xv.w * xv.w;
    }
#pragma unroll
    for (int off = 16; off > 0; off >>= 1) {
        s  += __shfl_xor(s,  off, 32);
        ss += __shfl_xor(ss, off, 32);
    }
    float mean = s * (1.f / 1024.f);
    float var  = ss * (1.f / 1024.f) - mean * mean;
    float rstd = rsqrtf(var + 1e-5f);

    s = 0.f; ss = 0.f;
#pragma unroll
    for (int i = 0; i < 8; ++i) {
        const int idx = i * 32 + lane;
        float4 gv = g14[idx], bv = b14[idx], fv = ff4[idx];
        float4 zv;
        zv.x = gv.x * (v[i].x - mean) * rstd + bv.x + fv.x;
        zv.y = gv.y * (v[i].y - mean) * rstd + bv.y + fv.y;
        zv.z = gv.z * (v[i].z - mean) * rstd + bv.z + fv.z;
        zv.w = gv.w * (v[i].w - mean) * rstd + bv.w + fv.w;
        v[i] = zv;
        s  += zv.x + zv.y + zv.z + zv.w;
        ss += zv.x * zv.x + zv.y * zv.y + zv.z * zv.z + zv.w * zv.w;
    }
#pragma unroll
    for (int off = 16; off > 0; off >>= 1) {
        s  += __shfl_xor(s,  off, 32);
        ss += __shfl_xor(ss, off, 32);
    }
    float mean2 = s * (1.f / 1024.f);
    float var2  = ss * (1.f / 1024.f) - mean2 * mean2;
    float rstd2 = rsqrtf(var2 + 1e-5f);

#pragma unroll
    for (int i = 0; i < 8; ++i) {
        const int idx = i * 32 + lane;
        float4 gv = g24[idx], bv = b24[idx];
        float4 ov;
        ov.x = gv.x * (v[i].x - mean2) * rstd2 + bv.x;
        ov.y = gv.y * (v[i].y - mean2) * rstd2 + bv.y;
        ov.z = gv.z * (v[i].z - mean2) * rstd2 + bv.z;
        ov.w = gv.w * (v[i].w - mean2) * rstd2 + bv.w;
        orow[idx] = ov;
    }
}

// ---------------------------------------------------------------------------
extern "C" void kernel_launch(void* const* d_in, const int* in_sizes, int n_in,
                              void* d_out, int out_size, void* d_ws, size_t ws_size,
                              hipStream_t stream) {
    (void)in_sizes; (void)n_in; (void)out_size; (void)ws_size;

    const float* x           = (const float*)d_in[0];
    const float* w_mix       = (const float*)d_in[4];
    const float* attn_params = (const float*)d_in[5];
    const float* w2          = (const float*)d_in[7];
    const float* w_out       = (const float*)d_in[8];
    const float* ffn_params  = (const float*)d_in[9];
    const float* g1          = (const float*)d_in[10];
    const float* b1          = (const float*)d_in[11];
    const float* g2          = (const float*)d_in[12];
    const float* b2          = (const float*)d_in[13];
    float*       out         = (float*)d_out;

    float* ws       = (float*)d_ws;
    float* hvec     = ws;            // 1024
    float* relu_vec = ws + 1024;     // 4096
    float* attn_out = ws + 5120;     // 1024
    float* ffn_out  = ws + 6144;     // 1024

    k_const_small<<<FFN / 256, 256, 0, stream>>>(attn_params, ffn_params, w2,
                                                 hvec, relu_vec);
    k_matvec_wmma<<<128, 32, 0, stream>>>(w_mix, w_out, hvec, relu_vec,
                                          attn_out, ffn_out);
    k_fused_ln<<<NTOK / 8, 256, 0, stream>>>(x, attn_out, ffn_out,
                                             g1, b1, g2, b2, out);
}